// GATProcessor_39642548142787
// MI455X (gfx1250) — compile-verified
//
#include <hip/hip_runtime.h>

typedef __attribute__((ext_vector_type(2))) float v2f;
typedef __attribute__((ext_vector_type(8))) float v8f;

#define NEG_SLOPE 0.2f
#define EPSV 1e-16f

// ---------- helpers ----------
__device__ __forceinline__ unsigned f32_ord(float f) {
    unsigned u = __float_as_uint(f);
    return ((int)u >= 0) ? (u | 0x80000000u) : ~u;   // order-preserving encode
}
__device__ __forceinline__ float ord_f32(unsigned u) {
    return __uint_as_float((u & 0x80000000u) ? (u & 0x7FFFFFFFu) : ~u);
}

// ---------- zero fill ----------
__global__ void fill_zero_u32(unsigned* __restrict__ p, size_t n) {
    size_t i = (size_t)blockIdx.x * blockDim.x + threadIdx.x;
    if (i < n) p[i] = 0u;
}

// ---------- GEMM: H[N,256] = X[N,64] @ Wf[64,256], fp32 WMMA 16x16x4 ----------
// one wave32 per 16x16 output tile; 16 k-steps of K=4
__global__ __launch_bounds__(256)
void gat_gemm_f32(const float* __restrict__ X, const float* __restrict__ Wf,
                  float* __restrict__ H, int M) {
    int wave = (int)((blockIdx.x * blockDim.x + threadIdx.x) >> 5);
    int lane = threadIdx.x & 31;
    const int nTilesN = 16;                 // 256 / 16
    int mt = wave / nTilesN;
    int ct = wave - mt * nTilesN;
    if (mt * 16 >= M) return;               // whole-wave uniform -> EXEC all-1 inside

    int lo  = lane & 15;
    int hi  = lane >> 4;
    int m   = mt * 16 + lo;
    int mrd = (m < M) ? m : (M - 1);        // clamp reads for partial tiles
    int col = ct * 16 + lo;

    const float* xrow = X + (size_t)mrd * 64;
    v8f c = {};
    #pragma unroll
    for (int t = 0; t < 16; ++t) {
        int k = t * 4 + hi * 2;             // A: VGPR0/1 = K {2*hi, 2*hi+1}
        v2f a;
        a.x = xrow[k];
        a.y = xrow[k + 1];
        v2f b;                              // B: rows K = 2*hi, 2*hi+1 ; col = lane&15
        b.x = Wf[(size_t)k * 256 + col];
        b.y = Wf[(size_t)(k + 1) * 256 + col];
        c = __builtin_amdgcn_wmma_f32_16x16x4_f32(false, a, false, b,
                                                  (short)0, c, false, false);
    }
    #pragma unroll
    for (int r = 0; r < 8; ++r) {           // C/D: VGPR r -> M = r + 8*hi
        int row = mt * 16 + r + hi * 8;
        if (row < M) H[(size_t)row * 256 + col] = c[r];
    }
}

// ---------- per-node attention logits: alpha_src/dst[N,4] ----------
// one wave per (node, head); lane covers 2 of the 64 features
__global__ __launch_bounds__(256)
void gat_alpha(const float* __restrict__ H,
               const float* __restrict__ a_src, const float* __restrict__ a_dst,
               float* __restrict__ asrc, float* __restrict__ adst, int N) {
    int wave = (int)((blockIdx.x * blockDim.x + threadIdx.x) >> 5);
    int lane = threadIdx.x & 31;
    int n  = wave >> 2;
    int hd = wave & 3;
    if (n >= N) return;
    const float* hr = H + (size_t)n * 256 + hd * 64;
    const float* as = a_src + hd * 64;
    const float* ad = a_dst + hd * 64;
    float h0 = hr[lane], h1 = hr[lane + 32];
    float s0 = h0 * as[lane] + h1 * as[lane + 32];
    float s1 = h0 * ad[lane] + h1 * ad[lane + 32];
    #pragma unroll
    for (int off = 16; off > 0; off >>= 1) {
        s0 += __shfl_down(s0, off, 32);
        s1 += __shfl_down(s1, off, 32);
    }
    if (lane == 0) {
        asrc[n * 4 + hd] = s0;
        adst[n * 4 + hd] = s1;
    }
}

// ---------- edge pass 1: segment max (ordered-uint atomicMax) ----------
__global__ __launch_bounds__(256)
void gat_edge_max(const long long* __restrict__ src, const long long* __restrict__ dst,
                  const float* __restrict__ asrc, const float* __restrict__ adst,
                  unsigned* __restrict__ menc, int E) {
    int e = (int)(blockIdx.x * blockDim.x + threadIdx.x);
    if (e >= E) return;
    int s = (int)src[e], d = (int)dst[e];
    #pragma unroll
    for (int h = 0; h < 4; ++h) {
        float v = asrc[s * 4 + h] + adst[d * 4 + h];
        v = (v > 0.f) ? v : NEG_SLOPE * v;
        atomicMax(&menc[d * 4 + h], f32_ord(v));
    }
}

// ---------- edge pass 2: segment exp-sum ----------
__global__ __launch_bounds__(256)
void gat_edge_sum(const long long* __restrict__ src, const long long* __restrict__ dst,
                  const float* __restrict__ asrc, const float* __restrict__ adst,
                  const unsigned* __restrict__ menc, float* __restrict__ ssum, int E) {
    int e = (int)(blockIdx.x * blockDim.x + threadIdx.x);
    if (e >= E) return;
    int s = (int)src[e], d = (int)dst[e];
    #pragma unroll
    for (int h = 0; h < 4; ++h) {
        float v = asrc[s * 4 + h] + adst[d * 4 + h];
        v = (v > 0.f) ? v : NEG_SLOPE * v;
        float ex = expf(v - ord_f32(menc[d * 4 + h]));
        atomicAdd(&ssum[d * 4 + h], ex);
    }
}

// ---------- edge pass 3: weighted aggregation with head-mean folded in ----------
// one wave per edge; lane handles 2 output columns; 4x fewer atomics than [N,H,O]
__global__ __launch_bounds__(256)
void gat_edge_aggr(const long long* __restrict__ src, const long long* __restrict__ dst,
                   const float* __restrict__ asrc, const float* __restrict__ adst,
                   const unsigned* __restrict__ menc, const float* __restrict__ ssum,
                   const float* __restrict__ H, float* __restrict__ acc, int E) {
    int wave = (int)((blockIdx.x * blockDim.x + threadIdx.x) >> 5);
    int lane = threadIdx.x & 31;
    if (wave >= E) return;
    int s = (int)src[wave], d = (int)dst[wave];
    float w[4];
    #pragma unroll
    for (int h = 0; h < 4; ++h) {
        float v = asrc[s * 4 + h] + adst[d * 4 + h];
        v = (v > 0.f) ? v : NEG_SLOPE * v;
        float ex = expf(v - ord_f32(menc[d * 4 + h]));
        w[h] = 0.25f * ex / (ssum[d * 4 + h] + EPSV);   // fold mean over 4 heads
    }
    const float* hr = H + (size_t)s * 256;
    #pragma unroll
    for (int rep = 0; rep < 2; ++rep) {
        int o = lane + rep * 32;
        float y = w[0] * hr[o] + w[1] * hr[64 + o] + w[2] * hr[128 + o] + w[3] * hr[192 + o];
        atomicAdd(&acc[(size_t)d * 64 + o], y);
    }
}

// ---------- finalize: out = relu(acc + bias) ----------
__global__ __launch_bounds__(256)
void gat_finalize(const float* __restrict__ acc, const float* __restrict__ bias,
                  float* __restrict__ out, int N) {
    int i = (int)(blockIdx.x * blockDim.x + threadIdx.x);
    if (i >= N * 64) return;
    float v = acc[i] + bias[i & 63];
    out[i] = (v > 0.f) ? v : 0.f;
}

// ---------- host side ----------
static inline size_t alignup(size_t v) { return (v + 255) & ~(size_t)255; }
static inline int cdiv(long long a, long long b) { return (int)((a + b - 1) / b); }

static void run_gat_layer(const float* xin, const float* W,
                          const float* a_src, const float* a_dst, const float* bias,
                          const long long* src, const long long* dst,
                          float* H, float* asrc, float* adst,
                          unsigned* menc, float* ssum, float* acc,
                          float* xout, int N, int E, hipStream_t stream) {
    // zero menc + ssum + acc (laid out contiguously: N*4 + N*4 + N*64 dwords)
    size_t zn = (size_t)N * 72;
    fill_zero_u32<<<cdiv((long long)zn, 256), 256, 0, stream>>>(menc, zn);

    int mtiles = cdiv(N, 16);
    int waves  = mtiles * 16;                      // 16 column tiles
    gat_gemm_f32<<<cdiv(waves, 8), 256, 0, stream>>>(xin, W, H, N);

    gat_alpha<<<cdiv((long long)N * 4, 8), 256, 0, stream>>>(H, a_src, a_dst, asrc, adst, N);

    gat_edge_max <<<cdiv(E, 256), 256, 0, stream>>>(src, dst, asrc, adst, menc, E);
    gat_edge_sum <<<cdiv(E, 256), 256, 0, stream>>>(src, dst, asrc, adst, menc, ssum, E);
    gat_edge_aggr<<<cdiv(E, 8),   256, 0, stream>>>(src, dst, asrc, adst, menc, ssum, H, acc, E);

    gat_finalize<<<cdiv((long long)N * 64, 256), 256, 0, stream>>>(acc, bias, xout, N);
}

extern "C" void kernel_launch(void* const* d_in, const int* in_sizes, int n_in,
                              void* d_out, int out_size, void* d_ws, size_t ws_size,
                              hipStream_t stream) {
    const float*     x  = (const float*)d_in[0];
    const long long* ei = (const long long*)d_in[1];   // int64 [2, E]
    // d_in[2] edge_attr unused; d_in[3] dropout_rate unused
    const float* W1  = (const float*)d_in[4];
    const float* as1 = (const float*)d_in[5];
    const float* ad1 = (const float*)d_in[6];
    const float* b1  = (const float*)d_in[7];
    const float* W2  = (const float*)d_in[8];
    const float* as2 = (const float*)d_in[9];
    const float* ad2 = (const float*)d_in[10];
    const float* b2  = (const float*)d_in[11];

    int N = in_sizes[0] / 64;
    int E = in_sizes[1] / 2;
    const long long* src = ei;
    const long long* dst = ei + E;

    char* ws = (char*)d_ws;
    float*    H    = (float*)ws;   ws += alignup((size_t)N * 256 * 4);
    float*    asrc = (float*)ws;   ws += alignup((size_t)N * 4 * 4);
    float*    adst = (float*)ws;   ws += alignup((size_t)N * 4 * 4);
    // menc, ssum, acc kept contiguous so one fill kernel zeroes all three
    unsigned* menc = (unsigned*)ws; ws += (size_t)N * 4 * 4;
    float*    ssum = (float*)ws;    ws += (size_t)N * 4 * 4;
    float*    acc  = (float*)ws;    ws += alignup((size_t)N * 64 * 4);
    float*    xmid = (float*)ws;    ws += alignup((size_t)N * 64 * 4);

    run_gat_layer(x,    W1, as1, ad1, b1, src, dst, H, asrc, adst, menc, ssum, acc,
                  xmid, N, E, stream);
    run_gat_layer(xmid, W2, as2, ad2, b2, src, dst, H, asrc, adst, menc, ssum, acc,
                  (float*)d_out, N, E, stream);
}